// LoraLinear_65738769433003
// MI455X (gfx1250) — compile-verified
//
#include <hip/hip_runtime.h>
#include <hip/hip_bf16.h>

// LoRA multi-adapter dispatch for MI455X (gfx1250, wave32, WMMA).
//
// out = result + sum_n (x . mask_n) @ A[n] @ B[n]   (rows adapter-exclusive)
//
// Stage 1 (shrink): V[T,16] = sum_n (X . mask_n) @ A[n], via masked
//   V_WMMA_F32_16X16X4_F32 chained into one accumulator per 16-token tile.
//   K (=H) split into SEGS segments for parallelism; deterministic partial
//   buffers in d_ws (no float atomics).
// Stage 2 (expand): out_tile = result_tile + sum_n (V . mask_n) @ B[n],
//   accumulator seeded with the result tile (fused add).

typedef float v2f __attribute__((ext_vector_type(2)));
typedef float v8f __attribute__((ext_vector_type(8)));

#define T_SZ   16384
#define H_SZ   4096
#define O_SZ   4096
#define NA     8
#define RANK_  16
#define NT     (T_SZ / 16)       // 1024 token tiles
#define SEGS   8
#define KSEG   (H_SZ / SEGS)     // 512
#define OT_PER_WAVE 4            // output tiles (of 16) per wave in expand

static __device__ __forceinline__ v8f wmma_f32(v2f a, v2f b, v8f c) {
    // 8 args: (neg_a, A, neg_b, B, c_mod, C, reuse_a, reuse_b)
    return __builtin_amdgcn_wmma_f32_16x16x4_f32(false, a, false, b,
                                                 (short)0, c, false, false);
}

// ---------------------------------------------------------------------------
// Stage 1: shrink.  One wave per (token-tile, K-segment).
// A-operand (16x4 f32): lane m = lane&15 -> token row, half = lane>>4:
//   V0 holds K = k0 + 2*half, V1 holds K = k0 + 2*half + 1.
// B-operand (4x16 f32): lane m -> rank column, same K split by half.
// C/D (16x16): VGPR i -> token (i + 8*half), col = m (rank).
// ---------------------------------------------------------------------------
__global__ __launch_bounds__(32)
void lora_shrink_kernel(const float* __restrict__ x,
                        const float* __restrict__ lora_a,   // [NA][H][RANK]
                        const int*   __restrict__ adapter_idx,
                        float*       __restrict__ vpart)    // [SEGS][NT][16][16]
{
    const int tile = blockIdx.x;          // token tile
    const int seg  = blockIdx.y;          // K segment
    const int lane = threadIdx.x;         // 0..31
    const int m    = lane & 15;
    const int half = lane >> 4;

    const int   token  = tile * 16 + m;
    const int   a_lane = adapter_idx[token];
    const float* xrow  = x + (size_t)token * H_SZ;

    v8f c = {};

    const int k_beg = seg * KSEG;
    const int k_end = k_beg + KSEG;
    for (int k0 = k_beg; k0 < k_end; k0 += 4) {
        // X chunk for this lane: two contiguous floats (8B-aligned -> b64 load)
        const v2f xa = *reinterpret_cast<const v2f*>(xrow + k0 + 2 * half);
#pragma unroll
        for (int n = 0; n < NA; ++n) {
            v2f am;
            am.x = (a_lane == n) ? xa.x : 0.0f;
            am.y = (a_lane == n) ? xa.y : 0.0f;
            // lora_a[n][k0 + 2*half + {0,1}][m]
            const float* aw = lora_a
                + ((size_t)n * H_SZ + (size_t)(k0 + 2 * half)) * RANK_ + m;
            v2f bw;
            bw.x = aw[0];
            bw.y = aw[RANK_];
            c = wmma_f32(am, bw, c);   // exclusive rows -> shared accumulator
        }
    }

    // Store partial V tile row-major [token-in-tile][rank]
    float* vp = vpart + ((size_t)seg * NT + tile) * 256;
#pragma unroll
    for (int i = 0; i < 8; ++i) {
        const int trow = i + 8 * half;
        vp[trow * 16 + m] = c[i];
    }
}

// ---------------------------------------------------------------------------
// Stage 2: expand.  One wave per (token-tile, group of OT_PER_WAVE out-tiles).
// ---------------------------------------------------------------------------
__global__ __launch_bounds__(32)
void lora_expand_kernel(const float* __restrict__ result,
                        const float* __restrict__ lora_b,     // [NA][RANK][O]
                        const int*   __restrict__ adapter_idx,
                        const float* __restrict__ vpart,      // [SEGS][NT][16][16]
                        float*       __restrict__ out)
{
    const int tile = blockIdx.x;          // token tile
    const int og   = blockIdx.y;          // output group
    const int lane = threadIdx.x;
    const int m    = lane & 15;
    const int half = lane >> 4;

    const int token  = tile * 16 + m;
    const int a_lane = adapter_idx[token];

    // Reduce the SEGS partial V tiles; this lane needs ranks
    // r = kc*4 + 2*half + {0,1} for kc = 0..3 of its token row m.
    float va[8];
#pragma unroll
    for (int kc = 0; kc < 4; ++kc) {
        const int r0 = kc * 4 + 2 * half;
        float s0 = 0.0f, s1 = 0.0f;
#pragma unroll
        for (int s = 0; s < SEGS; ++s) {
            const float* vp = vpart + ((size_t)s * NT + tile) * 256 + m * 16;
            s0 += vp[r0];
            s1 += vp[r0 + 1];
        }
        va[kc * 2 + 0] = s0;
        va[kc * 2 + 1] = s1;
    }

    // Pre-build the 8 masked A-operands (reused across output tiles).
    v2f am[NA][4];
#pragma unroll
    for (int n = 0; n < NA; ++n) {
#pragma unroll
        for (int kc = 0; kc < 4; ++kc) {
            am[n][kc].x = (a_lane == n) ? va[kc * 2 + 0] : 0.0f;
            am[n][kc].y = (a_lane == n) ? va[kc * 2 + 1] : 0.0f;
        }
    }

    for (int t = 0; t < OT_PER_WAVE; ++t) {
        const int o0 = (og * OT_PER_WAVE + t) * 16;

        // Seed accumulator with result tile (fused add). C/D layout:
        // VGPR i -> token (i + 8*half), column o0 + m.
        const float* rbase = result + (size_t)(tile * 16) * O_SZ + o0 + m;
        v8f c;
#pragma unroll
        for (int i = 0; i < 8; ++i)
            c[i] = rbase[(size_t)(i + 8 * half) * O_SZ];

#pragma unroll
        for (int n = 0; n < NA; ++n) {
#pragma unroll
            for (int kc = 0; kc < 4; ++kc) {
                // lora_b[n][kc*4 + 2*half + {0,1}][o0 + m]
                const float* bw = lora_b
                    + ((size_t)n * RANK_ + (size_t)(kc * 4 + 2 * half)) * O_SZ
                    + o0 + m;
                v2f b;
                b.x = bw[0];
                b.y = bw[O_SZ];
                c = wmma_f32(am[n][kc], b, c);
            }
        }

        float* obase = out + (size_t)(tile * 16) * O_SZ + o0 + m;
#pragma unroll
        for (int i = 0; i < 8; ++i)
            obase[(size_t)(i + 8 * half) * O_SZ] = c[i];
    }
}

extern "C" void kernel_launch(void* const* d_in, const int* in_sizes, int n_in,
                              void* d_out, int out_size, void* d_ws, size_t ws_size,
                              hipStream_t stream) {
    const float* result = (const float*)d_in[0];   // [T,O]
    const float* x      = (const float*)d_in[1];   // [T,H]
    const float* lora_a = (const float*)d_in[2];   // [NA,H,RANK]
    const float* lora_b = (const float*)d_in[3];   // [NA,RANK,O]
    const int*   a_idx  = (const int*)d_in[4];     // [T] (harness int)
    float*       out    = (float*)d_out;           // [T,O]
    float*       vpart  = (float*)d_ws;            // SEGS*NT*256*4 = 8 MB

    (void)in_sizes; (void)n_in; (void)out_size; (void)ws_size;

    dim3 blk(32, 1, 1);
    dim3 g1(NT, SEGS, 1);
    lora_shrink_kernel<<<g1, blk, 0, stream>>>(x, lora_a, a_idx, vpart);

    dim3 g2(NT, O_SZ / (16 * OT_PER_WAVE), 1);
    lora_expand_kernel<<<g2, blk, 0, stream>>>(result, lora_b, a_idx, vpart, out);
}